// Bert_entity_26242250178856
// MI455X (gfx1250) — compile-verified
//
#include <hip/hip_runtime.h>
#include <hip/hip_bf16.h>
#include <math.h>

#define Bn 64
#define Sn 512
#define Hn 768
#define Tn 9
#define KH 384   // half of H streamed through LDS at a time
#define LSTR (KH + 4)   // LDS row stride (floats): 388 % 64 == 4 -> conflict-free

typedef __attribute__((ext_vector_type(2))) float v2f;
typedef __attribute__((ext_vector_type(8))) float v8f;

// ---------------------------------------------------------------------------
// Kernel 1: per-batch stable compaction index (prefix sum over valid_ids)
// idxmap[b][j] = source position of j-th valid token, or -1 (zero row)
// ---------------------------------------------------------------------------
__global__ void compact_idx_kernel(const int* __restrict__ valid,
                                   int* __restrict__ idxmap,
                                   int* __restrict__ counts) {
    __shared__ int pre[Sn];
    __shared__ int sidx[Sn];
    const int b = blockIdx.x;
    const int t = threadIdx.x;
    const int v = valid[b * Sn + t];
    pre[t] = v;
    __syncthreads();
    // inclusive Hillis-Steele scan
    for (int off = 1; off < Sn; off <<= 1) {
        int x = pre[t];
        int y = (t >= off) ? pre[t - off] : 0;
        __syncthreads();
        pre[t] = x + y;
        __syncthreads();
    }
    const int total = pre[Sn - 1];
    sidx[t] = -1;
    __syncthreads();
    if (v) sidx[pre[t] - 1] = t;   // exclusive position
    __syncthreads();
    idxmap[b * Sn + t] = sidx[t];
    if (t == 0) counts[b] = total;
}

// ---------------------------------------------------------------------------
// Kernel 2: fused compaction-gather + classifier GEMM via V_WMMA_F32_16X16X4_F32
// One wave (32 threads) per 16-row tile. grid = B * (S/16) = 2048 blocks.
//   A = 16 x 768 tile of compacted hidden (FP32), streamed in 2 K-halves
//   B = cls_w^T slice, zero-padded to 16 rows IN LDS (unconditional ds_load_b64,
//       no EXEC-masked loads in the WMMA loop)
// Rows with idx == -1 give zero A rows -> logits = bias (matches reference).
// ---------------------------------------------------------------------------
__global__ void __launch_bounds__(32)
tile_gemm_kernel(const float* __restrict__ hidden,
                 const float* __restrict__ cls_w,
                 const float* __restrict__ cls_b,
                 const int* __restrict__ idxmap,
                 float* __restrict__ valid_out,
                 float* __restrict__ logits) {
    __shared__ float lA[16][LSTR];
    __shared__ float lW[16][LSTR];

    const int b    = blockIdx.x >> 5;       // 32 tiles per batch
    const int tile = blockIdx.x & 31;
    const int j0   = tile * 16;
    const int lane = threadIdx.x;           // 0..31, wave32
    const int m    = lane & 15;
    const int hsel = lane >> 4;             // 0: K+0/1, 1: K+2/3
    const int co   = 2 * hsel;

    v8f c = {};
    for (int halfk = 0; halfk < 2; ++halfk) {
        const int kbase = halfk * KH;

        // ---- stage weight tile: rows 0..8 from cls_w, rows 9..15 zeros ----
        for (int t = 0; t < 16; ++t) {
            if (t < Tn) {
                const float* wp = cls_w + t * Hn + kbase;
                for (int i = lane * 4; i < KH; i += 128)
                    *(float4*)&lW[t][i] = *(const float4*)(wp + i);
            } else {
                const float4 z4 = make_float4(0.f, 0.f, 0.f, 0.f);
                for (int i = lane * 4; i < KH; i += 128)
                    *(float4*)&lW[t][i] = z4;
            }
        }

        // ---- stage A tile (compacted rows); fuse valid_output store ----
        for (int r = 0; r < 16; ++r) {
            const int src = idxmap[b * Sn + j0 + r];     // wave-uniform
            float* vo = valid_out + ((size_t)(b * Sn + j0 + r)) * Hn + kbase;
            if (src >= 0) {
                const float* hp = hidden + ((size_t)b * Sn + src) * Hn + kbase;
                for (int i = lane * 4; i < KH; i += 128) {
                    float4 v4 = *(const float4*)(hp + i);
                    *(float4*)&lA[r][i] = v4;
                    // d_out-relative base is only 4B aligned -> scalar stores
                    vo[i]     = v4.x;
                    vo[i + 1] = v4.y;
                    vo[i + 2] = v4.z;
                    vo[i + 3] = v4.w;
                }
            } else {
                for (int i = lane * 4; i < KH; i += 128) {
                    *(float4*)&lA[r][i] = make_float4(0.f, 0.f, 0.f, 0.f);
                    vo[i] = 0.f; vo[i + 1] = 0.f; vo[i + 2] = 0.f; vo[i + 3] = 0.f;
                }
            }
        }
        __syncthreads();

        // ---- chained 16x16x4 FP32 WMMAs over this K-half ----
#pragma unroll 4
        for (int k = 0; k < KH; k += 4) {
            v2f a, bb;
            a.x  = lA[m][k + co];
            a.y  = lA[m][k + co + 1];
            bb.x = lW[m][k + co];
            bb.y = lW[m][k + co + 1];
            c = __builtin_amdgcn_wmma_f32_16x16x4_f32(
                    /*neg_a=*/false, a, /*neg_b=*/false, bb,
                    /*c_mod=*/(short)0, c, /*reuse_a=*/false, /*reuse_b=*/false);
        }
        __syncthreads();
    }

    // C/D layout: VGPR r holds row M=r (lanes 0-15) and M=r+8 (lanes 16-31), N=lane&15
    if (m < Tn) {
        const float bias = cls_b[m];
        for (int r = 0; r < 8; ++r) {
            const int row = r + 8 * hsel;
            logits[((size_t)(b * Sn + j0 + row)) * Tn + m] = c[r] + bias;
        }
    }
}

// ---------------------------------------------------------------------------
// Kernel 3: CRF log-likelihood terms: batch_loss[b] = score - logZ
// One wave per batch; lanes 0..8 carry alpha states; lane 0 does scalar parts.
// ---------------------------------------------------------------------------
__global__ void crf_loss_kernel(const float* __restrict__ logits,
                                const int* __restrict__ labels,
                                const int* __restrict__ counts,
                                const float* __restrict__ start,
                                const float* __restrict__ endt,
                                const float* __restrict__ trans,
                                float* __restrict__ batch_loss) {
    __shared__ float sh_alpha[Tn];
    __shared__ float sh_trans[Tn * Tn];
    const int b = blockIdx.x;
    const int lane = threadIdx.x;
    const int cnt = counts[b];
    const float* lg = logits + (size_t)b * Sn * Tn;

    for (int i = lane; i < Tn * Tn; i += 32) sh_trans[i] = trans[i];
    if (lane < Tn) sh_alpha[lane] = start[lane] + lg[lane];
    __syncthreads();

    for (int s = 1; s < Sn; ++s) {
        float nxt = 0.0f;
        if (lane < Tn) {
            float mx = -__builtin_inff();
            float v[Tn];
            for (int i = 0; i < Tn; ++i) {
                v[i] = sh_alpha[i] + sh_trans[i * Tn + lane];
                mx = fmaxf(mx, v[i]);
            }
            float ssum = 0.0f;
            for (int i = 0; i < Tn; ++i) ssum += expf(v[i] - mx);
            nxt = mx + logf(ssum) + lg[s * Tn + lane];
        }
        __syncthreads();
        if (lane < Tn && s < cnt) sh_alpha[lane] = nxt;
        __syncthreads();
    }

    if (lane == 0) {
        // logZ = logsumexp(alpha + end)
        float mx = -__builtin_inff();
        for (int j = 0; j < Tn; ++j) mx = fmaxf(mx, sh_alpha[j] + endt[j]);
        float ssum = 0.0f;
        for (int j = 0; j < Tn; ++j) ssum += expf(sh_alpha[j] + endt[j] - mx);
        const float logZ = mx + logf(ssum);
        // gold path score
        const int* tg = labels + b * Sn;
        float score = start[tg[0]] + lg[tg[0]];
        for (int s = 1; s < cnt; ++s)
            score += sh_trans[tg[s - 1] * Tn + tg[s]] + lg[s * Tn + tg[s]];
        score += endt[tg[cnt - 1]];
        batch_loss[b] = score - logZ;
    }
}

// ---------------------------------------------------------------------------
// Kernel 4: Viterbi decode. History in ws; serial backtrack; masked float out.
// ---------------------------------------------------------------------------
__global__ void viterbi_kernel(const float* __restrict__ logits,
                               const int* __restrict__ counts,
                               const float* __restrict__ start,
                               const float* __restrict__ endt,
                               const float* __restrict__ trans,
                               int* __restrict__ hist_ws,
                               float* __restrict__ decoded) {
    __shared__ float sh_alpha[Tn];
    __shared__ float sh_trans[Tn * Tn];
    __shared__ int dec[Sn];
    const int b = blockIdx.x;
    const int lane = threadIdx.x;
    const int cnt = counts[b];
    const float* lg = logits + (size_t)b * Sn * Tn;
    int* hist = hist_ws + (size_t)b * (Sn - 1) * Tn;

    for (int i = lane; i < Tn * Tn; i += 32) sh_trans[i] = trans[i];
    if (lane < Tn) sh_alpha[lane] = start[lane] + lg[lane];
    __syncthreads();

    for (int s = 1; s < Sn; ++s) {
        float nxt = 0.0f;
        int idx = lane;
        if (lane < Tn) {
            float best = -__builtin_inff();
            int bi = 0;
            for (int i = 0; i < Tn; ++i) {
                float v = sh_alpha[i] + sh_trans[i * Tn + lane];
                if (v > best) { best = v; bi = i; }   // first-max tie like argmax
            }
            nxt = best + lg[s * Tn + lane];
            idx = (s < cnt) ? bi : lane;              // masked step: identity
        }
        __syncthreads();
        if (lane < Tn) {
            if (s < cnt) sh_alpha[lane] = nxt;
            hist[(s - 1) * Tn + lane] = idx;
        }
        __syncthreads();
    }

    if (lane == 0) {
        float best = -__builtin_inff();
        int tag = 0;
        for (int j = 0; j < Tn; ++j) {
            float v = sh_alpha[j] + endt[j];
            if (v > best) { best = v; tag = j; }
        }
        dec[Sn - 1] = tag;
        for (int s = Sn - 1; s >= 1; --s) {
            tag = hist[(s - 1) * Tn + tag];
            dec[s - 1] = tag;
        }
    }
    __syncthreads();
    for (int s = lane; s < Sn; s += 32)
        decoded[b * Sn + s] = (s < cnt) ? (float)dec[s] : 0.0f;
}

// ---------------------------------------------------------------------------
// Kernel 5: loss = sum(batch_loss) / sum(counts)
// ---------------------------------------------------------------------------
__global__ void loss_reduce_kernel(const float* __restrict__ batch_loss,
                                   const int* __restrict__ counts,
                                   float* __restrict__ out_loss) {
    __shared__ float sl[Bn];
    __shared__ int sc[Bn];
    const int t = threadIdx.x;
    sl[t] = batch_loss[t];
    sc[t] = counts[t];
    __syncthreads();
    if (t == 0) {
        float s = 0.0f;
        int c = 0;
        for (int i = 0; i < Bn; ++i) { s += sl[i]; c += sc[i]; }
        out_loss[0] = s / (float)c;
    }
}

// ---------------------------------------------------------------------------
extern "C" void kernel_launch(void* const* d_in, const int* in_sizes, int n_in,
                              void* d_out, int out_size, void* d_ws, size_t ws_size,
                              hipStream_t stream) {
    const float* hidden = (const float*)d_in[0];   // (B,S,H) f32
    const int*   valid  = (const int*)  d_in[1];   // (B,S)   i32
    const int*   labels = (const int*)  d_in[2];   // (B,S)   i32
    const float* cls_w  = (const float*)d_in[3];   // (T,H)
    const float* cls_b  = (const float*)d_in[4];   // (T,)
    const float* start  = (const float*)d_in[5];   // (T,)
    const float* endt   = (const float*)d_in[6];   // (T,)
    const float* trans  = (const float*)d_in[7];   // (T,T)

    // output: [loss(1)] [decoded(B*S)] [valid_output(B*S*H)] as floats
    float* out_loss  = (float*)d_out;
    float* decoded   = out_loss + 1;
    float* valid_out = decoded + Bn * Sn;

    // workspace layout (256B-aligned regions)
    char* ws = (char*)d_ws;
    int*   idxmap     = (int*)(ws + 0);                       // B*S ints     (131072 B)
    int*   counts     = (int*)(ws + 131072);                  // B ints       (256 B)
    float* batch_loss = (float*)(ws + 131328);                // B floats     (256 B)
    int*   hist       = (int*)(ws + 131584);                  // B*511*T ints (1177344 B)
    float* logits     = (float*)(ws + 1308928);               // B*S*T floats (1179648 B)

    compact_idx_kernel<<<Bn, Sn, 0, stream>>>(valid, idxmap, counts);
    tile_gemm_kernel<<<Bn * (Sn / 16), 32, 0, stream>>>(hidden, cls_w, cls_b,
                                                        idxmap, valid_out, logits);
    crf_loss_kernel<<<Bn, 32, 0, stream>>>(logits, labels, counts, start, endt,
                                           trans, batch_loss);
    viterbi_kernel<<<Bn, 32, 0, stream>>>(logits, counts, start, endt, trans,
                                          hist, decoded);
    loss_reduce_kernel<<<1, Bn, 0, stream>>>(batch_loss, counts, out_loss);
}